// RankingLossL1_17746804867499
// MI455X (gfx1250) — compile-verified
//
#include <hip/hip_runtime.h>

// RankingLossL1 forward, MI455X (gfx1250, wave32).
// B = -dist at top-k indices, so only the K smallest L1 distance VALUES per
// anchor are needed. Per (anchor, direction) workgroup:
//   1) stream base matrix through LDS tiles via global_load_async_to_lds_b128
//      (double buffered, ASYNCcnt), computing all N distances into LDS
//   2) exact rank-(K-1) statistic via 4-pass radix select on float bits,
//      with a parallel 256-bin prefix-scan per pass (no serial thread-0 scan)
//   3) contribution = sum_{d<T} relu(D-d) + (K-cntLess)*relu(D-T)
// Deterministic: per-block partials in d_ws, fixed-order final reduction.

#define THREADS   256
#define KNEG      50
#define GAMMAC    1.0f
#define DIM       128
#define NT        64          // base rows per tile buffer
#define ROWSTRIDE 144         // floats per LDS tile row (576B): bank-conflict-free
#define MAXN      20032       // dist array capacity (>= N, mult of NT)

__device__ __forceinline__ unsigned lds_off(const void* p) {
  // generic pointer to LDS: low 32 bits are the LDS byte offset
  return (unsigned)(size_t)p;
}

__global__ __launch_bounds__(THREADS)
void rank_loss_main(const float* __restrict__ out1,
                    const float* __restrict__ out2,
                    const int*   __restrict__ anchor1,
                    const int*   __restrict__ anchor2,
                    float*       __restrict__ partial,
                    int N) {
  __shared__ __align__(16) float tile[2][NT * ROWSTRIDE]; // 73728 B
  __shared__ __align__(16) float dist[MAXN];              // 80128 B
  __shared__ __align__(16) float qv[DIM];
  __shared__ unsigned hist[256];
  __shared__ float red[8];
  __shared__ float shD;
  __shared__ unsigned shPref;
  __shared__ int shRem, shLess;

  const int tid = threadIdx.x;
  const int blk = blockIdx.x;
  const int a   = blk >> 1;
  const int dir = blk & 1;

  const int i1 = anchor1[a];
  const int i2 = anchor2[a];
  const float* __restrict__ base = dir ? out1 : out2;

  // ---- pos = L1(ae1, ae2); stage query vector in LDS ----
  float pd = 0.0f;
  if (tid < DIM) {
    float e1 = out1[(size_t)i1 * DIM + tid];
    float e2 = out2[(size_t)i2 * DIM + tid];
    qv[tid] = dir ? e2 : e1;
    pd = fabsf(e1 - e2);
  }
#pragma unroll
  for (int m = 16; m; m >>= 1) pd += __shfl_xor(pd, m);
  if ((tid & 31) == 0) red[tid >> 5] = pd;
  __syncthreads();
  if (tid == 0) {
    float pos = 0.f;
#pragma unroll
    for (int w = 0; w < 8; ++w) pos += red[w];
    shD = pos + GAMMAC;
  }
  __syncthreads();
  const float Dv = shD;

  // 4 threads cooperate per base row; each handles interleaved 16B chunks
  const int rowT = tid >> 2;    // 0..63
  const int sub  = tid & 3;     // chunk phase

  float4 qr[8];
#pragma unroll
  for (int k = 0; k < 8; ++k) {
    int chunk = k * 4 + sub;    // 16B chunk 0..31
    qr[k] = *reinterpret_cast<const float4*>(&qv[chunk * 4]);
  }

  const int numTiles = (N + NT - 1) / NT;

  auto load_tile = [&](int t, int bufsel) {
    const int r0 = t * NT;
#pragma unroll
    for (int k = 0; k < 8; ++k) {
      int c    = (k << 8) + tid;  // chunk id 0..2047 within the tile
      int row  = c >> 5;          // 0..63
      int kk   = c & 31;          // 16B chunk within row
      int grow = r0 + row;
      if (grow >= N) grow = N - 1;                 // clamp (padding rows)
      unsigned goff = (unsigned)(grow * (DIM * 4) + kk * 16);
      unsigned lds  = lds_off(&tile[bufsel][row * ROWSTRIDE + kk * 4]);
      asm volatile("global_load_async_to_lds_b128 %0, %1, %2"
                   :: "v"(lds), "v"(goff), "s"(base) : "memory");
    }
  };

  load_tile(0, 0);
  for (int t = 0; t < numTiles; ++t) {
    const int cur = t & 1;
    asm volatile("s_wait_asynccnt 0" ::: "memory");  // this wave's async loads done
    __syncthreads();                                 // all waves' loads + prev compute done
    if (t + 1 < numTiles) load_tile(t + 1, cur ^ 1); // prefetch next tile (other buffer)

    const int j = t * NT + rowT;
    float acc = 0.f;
#pragma unroll
    for (int k = 0; k < 8; ++k) {
      int chunk = k * 4 + sub;
      float4 b  = *reinterpret_cast<const float4*>(&tile[cur][rowT * ROWSTRIDE + chunk * 4]);
      float4 aq = qr[k];
      acc += fabsf(aq.x - b.x) + fabsf(aq.y - b.y) +
             fabsf(aq.z - b.z) + fabsf(aq.w - b.w);
    }
    acc += __shfl_xor(acc, 1);
    acc += __shfl_xor(acc, 2);
    if (sub == 0 && j < N) dist[j] = acc;
  }
  __syncthreads();

  // ---- exact rank-(K-1) selection: radix select over float bit patterns ----
  const int nvec = N & ~3;
  unsigned pref = 0u, himask = 0u;
  int rem = KNEG - 1, less = 0;
  for (int byt = 3; byt >= 0; --byt) {
    hist[tid] = 0u;
    __syncthreads();
    const int shift = byt * 8;
    // vectorized histogram sweep (float4, bank-conflict-free)
    for (int j = tid * 4; j < nvec; j += THREADS * 4) {
      float4 dv = *reinterpret_cast<const float4*>(&dist[j]);
      unsigned u;
      u = __float_as_uint(dv.x); if ((u & himask) == pref) atomicAdd(&hist[(u >> shift) & 255u], 1u);
      u = __float_as_uint(dv.y); if ((u & himask) == pref) atomicAdd(&hist[(u >> shift) & 255u], 1u);
      u = __float_as_uint(dv.z); if ((u & himask) == pref) atomicAdd(&hist[(u >> shift) & 255u], 1u);
      u = __float_as_uint(dv.w); if ((u & himask) == pref) atomicAdd(&hist[(u >> shift) & 255u], 1u);
    }
    for (int j = nvec + tid; j < N; j += THREADS) {
      unsigned u = __float_as_uint(dist[j]);
      if ((u & himask) == pref) atomicAdd(&hist[(u >> shift) & 255u], 1u);
    }
    __syncthreads();

    // parallel 256-bin inclusive prefix scan, in place in hist
    const unsigned h = hist[tid];
#pragma unroll
    for (int off = 1; off < 256; off <<= 1) {
      unsigned add = (tid >= off) ? hist[tid - off] : 0u;
      __syncthreads();
      hist[tid] += add;
      __syncthreads();
    }
    const unsigned inc = hist[tid];
    const unsigned exc = inc - h;
    if ((int)inc > rem && (int)exc <= rem) {   // exactly one thread: bin holding the rank
      shPref = pref | ((unsigned)tid << shift);
      shRem  = rem - (int)exc;
      shLess = less + (int)exc;
    }
    __syncthreads();
    pref = shPref; rem = shRem; less = shLess;
    himask |= (0xFFu << shift);
  }

  // ---- loss contribution ----
  const float T = __uint_as_float(pref);   // K-th smallest distance
  float s = 0.f;
  for (int j = tid * 4; j < nvec; j += THREADS * 4) {
    float4 dv = *reinterpret_cast<const float4*>(&dist[j]);
    if (dv.x < T) { float r = Dv - dv.x; s += (r > 0.f) ? r : 0.f; }
    if (dv.y < T) { float r = Dv - dv.y; s += (r > 0.f) ? r : 0.f; }
    if (dv.z < T) { float r = Dv - dv.z; s += (r > 0.f) ? r : 0.f; }
    if (dv.w < T) { float r = Dv - dv.w; s += (r > 0.f) ? r : 0.f; }
  }
  for (int j = nvec + tid; j < N; j += THREADS) {
    float dj = dist[j];
    if (dj < T) { float r = Dv - dj; s += (r > 0.f) ? r : 0.f; }
  }
#pragma unroll
  for (int m = 16; m; m >>= 1) s += __shfl_xor(s, m);
  if ((tid & 31) == 0) red[tid >> 5] = s;
  __syncthreads();
  if (tid == 0) {
    float tot = 0.f;
#pragma unroll
    for (int w = 0; w < 8; ++w) tot += red[w];
    float rT = Dv - T;
    tot += (float)(KNEG - less) * ((rT > 0.f) ? rT : 0.f);  // boundary ties
    partial[blk] = tot;
  }
}

__global__ __launch_bounds__(256)
void rank_loss_reduce(const float* __restrict__ partial,
                      float* __restrict__ out,
                      int nPartial, float invScale) {
  __shared__ float red[8];
  const int tid = threadIdx.x;
  float s = 0.f;
  for (int j = tid; j < nPartial; j += 256) s += partial[j];  // fixed order
#pragma unroll
  for (int m = 16; m; m >>= 1) s += __shfl_xor(s, m);
  if ((tid & 31) == 0) red[tid >> 5] = s;
  __syncthreads();
  if (tid == 0) {
    float t = 0.f;
#pragma unroll
    for (int w = 0; w < 8; ++w) t += red[w];
    out[0] = t * invScale;
  }
}

extern "C" void kernel_launch(void* const* d_in, const int* in_sizes, int n_in,
                              void* d_out, int out_size, void* d_ws, size_t ws_size,
                              hipStream_t stream) {
  const float* out1    = (const float*)d_in[0];
  const float* out2    = (const float*)d_in[1];
  const int*   anchor1 = (const int*)d_in[2];
  const int*   anchor2 = (const int*)d_in[3];
  const int A = in_sizes[2];
  const int N = in_sizes[0] / DIM;

  float* partial = (float*)d_ws;          // 2*A floats of scratch
  const int nblk = 2 * A;

  rank_loss_main<<<nblk, THREADS, 0, stream>>>(out1, out2, anchor1, anchor2,
                                               partial, N);
  const float invScale = 1.0f / ((float)A * (float)KNEG);
  rank_loss_reduce<<<1, 256, 0, stream>>>(partial, (float*)d_out, nblk, invScale);
}